// Cache_25142738551024
// MI455X (gfx1250) — compile-verified
//
#include <hip/hip_runtime.h>

// Sliding-window KV cache append for MI455X (gfx1250), implemented entirely
// with the CDNA5 Tensor Data Mover (TDM).
//
// Pure data movement: ~1.07 GB of HBM traffic, 0 FLOPs -> HBM-roofline bound
// (>= ~46 us at 23.3 TB/s). TDM DMAs contiguous chunks global->LDS->global,
// descriptors live in SGPRs, waves only build D# groups and wait on TENSORcnt.

typedef unsigned int u32x4 __attribute__((ext_vector_type(4)));
typedef int          i32x8 __attribute__((ext_vector_type(8)));
typedef int          i32x4 __attribute__((ext_vector_type(4)));

#define ROW_BYTES    512u                      // D=128 floats
#define PLANE_BYTES  (4096u * ROW_BYTES)       // 2,097,152 B per (b,h) plane
#define SKIP_BYTES   (16u * ROW_BYTES)         // 8,192 B (16 dropped rows)
#define BULK_BYTES   (4080u * ROW_BYTES)       // 2,088,960 B kept per plane
#define CHUNK_UNITS  8160u                     // dwords per chunk (fits tile_dim0)
#define CHUNK_BYTES  (CHUNK_UNITS * 4u)        // 32,640 B
#define CHUNKS_PER_PLANE 64u                   // 64 * 32,640 = 2,088,960
#define SUBS_PER_PLANE   4u                    // blocks per plane
#define CHUNKS_PER_SUB   (CHUNKS_PER_PLANE / SUBS_PER_PLANE)   // 16
#define TAIL_UNITS   2048u                     // 8,192 B of new tokens

// Build a 1-D TDM descriptor: copy `units` dwords between global `gaddr` and
// LDS byte offset `lds_addr`.  Layout per CDNA5 ISA.
static __device__ __forceinline__ void make_desc(u32x4 &g0, i32x8 &g1,
                                                 unsigned long long gaddr,
                                                 unsigned lds_addr,
                                                 unsigned units) {
  g0[0] = 1u;                                            // count=1 (valid), user mode
  g0[1] = lds_addr;                                      // lds_addr [63:32]
  g0[2] = (unsigned)(gaddr & 0xffffffffull);             // global_addr lo
  g0[3] = (unsigned)((gaddr >> 32) & 0x01ffffffull)      // global_addr [56:32]
        | (2u << 30);                                    // type = 2 ("image")
  g1[0] = (int)(2u << 16);                               // data_size = 4 B; no mcast/pad/iterate
  g1[1] = (int)((units & 0xffffu) << 16);                // tensor_dim0[15:0]
  g1[2] = (int)(((units >> 16) & 0xffffu) | (1u << 16)); // tensor_dim0[31:16] | tensor_dim1 = 1
  g1[3] = (int)((units & 0xffffu) << 16);                // tile_dim0 = units (1-D tile)
  g1[4] = 1;                                             // tile_dim1 = 1, tile_dim2 = 0
  g1[5] = (int)units;                                    // tensor_dim0_stride lo
  g1[6] = 0;                                             // stride0 hi | stride1 lo
  g1[7] = 0;                                             // stride1 hi
}

static __device__ __forceinline__ void tdm_load(unsigned long long src,
                                                unsigned lds, unsigned units,
                                                const i32x4 &gz4,
                                                const i32x8 &gz8) {
  u32x4 g0; i32x8 g1;
  make_desc(g0, g1, src, lds, units);
  __builtin_amdgcn_tensor_load_to_lds(g0, g1, gz4, gz4, gz8, 0);
}

static __device__ __forceinline__ void tdm_store(unsigned long long dst,
                                                 unsigned lds, unsigned units,
                                                 const i32x4 &gz4,
                                                 const i32x8 &gz8) {
  u32x4 g0; i32x8 g1;
  make_desc(g0, g1, dst, lds, units);
  __builtin_amdgcn_tensor_store_from_lds(g0, g1, gz4, gz4, gz8, 0);
}

__global__ void kv_append_tdm(const float* __restrict__ cache_k,
                              const float* __restrict__ cache_v,
                              const float* __restrict__ knew,
                              const float* __restrict__ vnew,
                              float* __restrict__ out) {
  __shared__ unsigned char lds[2][CHUNK_BYTES];          // 65,280 B double buffer

  const unsigned blk   = blockIdx.x;                     // 0..1023
  const unsigned sub   = blk & (SUBS_PER_PLANE - 1u);    // sub-range within plane
  const unsigned p     = blk / SUBS_PER_PLANE;           // plane 0..255
  const unsigned t     = p >> 7;                         // 0 = K, 1 = V
  const unsigned bh    = p & 127u;                       // (b*32 + h)

  const float* cache = t ? cache_v : cache_k;
  const float* fresh = t ? vnew    : knew;

  const unsigned long long srcPlane =
      (unsigned long long)reinterpret_cast<size_t>(cache)
      + (unsigned long long)bh * PLANE_BYTES + SKIP_BYTES;
  const unsigned long long newPlane =
      (unsigned long long)reinterpret_cast<size_t>(fresh)
      + (unsigned long long)bh * SKIP_BYTES;
  const unsigned long long dstPlane =
      (unsigned long long)reinterpret_cast<size_t>(out)
      + (unsigned long long)t * (128ull * PLANE_BYTES)
      + (unsigned long long)bh * PLANE_BYTES;

  const unsigned long long subOff =
      (unsigned long long)sub * CHUNKS_PER_SUB * CHUNK_BYTES;
  const unsigned long long srcBase = srcPlane + subOff;
  const unsigned long long dstBase = dstPlane + subOff;

  const unsigned ldsA = (unsigned)(unsigned long long)
                        reinterpret_cast<size_t>(&lds[0][0]);
  const unsigned ldsB = (unsigned)(unsigned long long)
                        reinterpret_cast<size_t>(&lds[1][0]);
  const i32x4 gz4 = {0, 0, 0, 0};
  const i32x8 gz8 = {0, 0, 0, 0, 0, 0, 0, 0};

  // Software-pipelined ping-pong: load(c+1) overlaps store(c).
  tdm_load(srcBase, ldsA, CHUNK_UNITS, gz4, gz8);
  for (unsigned c = 0; c < CHUNKS_PER_SUB; ++c) {
    const unsigned buf = (c & 1u) ? ldsB : ldsA;
    __builtin_amdgcn_s_wait_tensorcnt((short)0);         // chunk c in LDS, prior store done
    tdm_store(dstBase + (unsigned long long)c * CHUNK_BYTES, buf, CHUNK_UNITS,
              gz4, gz8);
    if (c + 1u < CHUNKS_PER_SUB) {
      const unsigned nbuf = ((c + 1u) & 1u) ? ldsB : ldsA;
      tdm_load(srcBase + (unsigned long long)(c + 1u) * CHUNK_BYTES,
               nbuf, CHUNK_UNITS, gz4, gz8);             // overlaps store(c)
    }
  }

  // Last sub-block of each plane appends the 16 new token rows (8 KiB).
  if (sub == SUBS_PER_PLANE - 1u) {
    __builtin_amdgcn_s_wait_tensorcnt((short)0);
    tdm_load(newPlane, ldsA, TAIL_UNITS, gz4, gz8);
    __builtin_amdgcn_s_wait_tensorcnt((short)0);
    tdm_store(dstPlane + BULK_BYTES, ldsA, TAIL_UNITS, gz4, gz8);
  }
  // Final outstanding store is covered by the implicit wait-idle at s_endpgm.
}

extern "C" void kernel_launch(void* const* d_in, const int* in_sizes, int n_in,
                              void* d_out, int out_size, void* d_ws, size_t ws_size,
                              hipStream_t stream) {
  (void)in_sizes; (void)n_in; (void)out_size; (void)d_ws; (void)ws_size;
  const float* cache_k = (const float*)d_in[0];
  const float* cache_v = (const float*)d_in[1];
  const float* knew    = (const float*)d_in[2];
  const float* vnew    = (const float*)d_in[3];
  float* out = (float*)d_out;

  // 256 planes x 4 sub-blocks, one wave32 per block (TDM is issued per-wave).
  dim3 grid(2u * 128u * SUBS_PER_PLANE);   // 1024 blocks
  dim3 block(32);
  kv_append_tdm<<<grid, block, 0, stream>>>(cache_k, cache_v, knew, vnew, out);
}